// BlockWiseEmbeddingForInput_58806692216985
// MI455X (gfx1250) — compile-verified
//
#include <hip/hip_runtime.h>

// BlockWiseEmbedding forward for MI455X (gfx1250, wave32).
//
// Roofline: 419 MB of f32 output stores dominate (~18us @ 23.3 TB/s); all
// embedding tables fit in the 192 MB L2 so gathers are cache-resident. The
// ~2.5 GFLOP of projection work would need ~140 TFLOPS f32 to hide at the
// store bound -> offload it to the matrix pipe with V_WMMA_F32_16X16X4_F32
// (full f32 precision, matches the f32 reference; no bf16 downcast).

typedef __attribute__((ext_vector_type(2))) float v2f;
typedef __attribute__((ext_vector_type(8))) float v8f;

#define EMBED_DIM 128
#define B0_HI 64000
#define B1_HI 256000
#define D1 32
#define D2 8
#define TOKENS (4096 * 200)   // 819200, divisible by 16

__global__ __launch_bounds__(256) void blockwise_embed_wmma(
    const int*   __restrict__ inputs,       // [B*S]
    const float* __restrict__ firstblock_w, // [64000, 128]
    const float* __restrict__ emb1,         // [192000, 32]
    const float* __restrict__ proj1,        // [32, 128]
    const float* __restrict__ emb2,         // [744000, 8]
    const float* __restrict__ proj2,        // [8, 128]
    float*       __restrict__ out)          // [B*S, 128]
{
    const int lane  = threadIdx.x & 31;
    const int wave  = threadIdx.x >> 5;
    const int tile  = blockIdx.x * 8 + wave;     // 16 tokens per wave
    const int tokenBase = tile * 16;
    if (tokenBase >= TOKENS) return;             // wave-uniform

    // A-fragment row mapping for V_WMMA_F32_16X16X4_F32:
    //   lanes 0-15  : M = lane,    K-pair {4k+0, 4k+1}
    //   lanes 16-31 : M = lane-16, K-pair {4k+2, 4k+3}
    const int m     = lane & 15;                 // matrix row owned for A, col for B/C
    const int khalf = (lane >> 4) << 1;          // 0 or 2: K offset within a chunk

    const int  raw = inputs[tokenBase + m];
    const bool in1 = (raw >= B0_HI) & (raw < B1_HI);
    const bool in2 = (raw >= B1_HI);
    const float m1 = in1 ? 1.0f : 0.0f;
    const float m2 = in2 ? 1.0f : 0.0f;
    const int  r1  = in1 ? (raw - B0_HI) : 0;    // masked rows read row 0 (valid)
    const int  r2  = in2 ? (raw - B1_HI) : 0;

    // ---- Gather + mask A fragments (reused across all 8 N-tiles) ----
    const float* e1p = emb1 + (size_t)r1 * D1;
    v2f a1[8];
#pragma unroll
    for (int k = 0; k < 8; ++k) {
        v2f v = *(const v2f*)(e1p + 4 * k + khalf);
        a1[k].x = v.x * m1;
        a1[k].y = v.y * m1;
    }
    const float* e2p = emb2 + (size_t)r2 * D2;
    v2f a2[2];
#pragma unroll
    for (int k = 0; k < 2; ++k) {
        v2f v = *(const v2f*)(e2p + 4 * k + khalf);
        a2[k].x = v.x * m2;
        a2[k].y = v.y * m2;
    }

    // ---- 8 N-tiles of 16 output columns each ----
#pragma unroll 1
    for (int n = 0; n < 8; ++n) {
        const int ncol = n * 16 + m;             // column this lane owns in B/C

        v8f c = {};
        // block 1: K = 32 -> 8 chunks of 4
#pragma unroll
        for (int k = 0; k < 8; ++k) {
            const int krow = 4 * k + khalf;
            v2f b;
            b.x = proj1[(size_t)krow * EMBED_DIM + ncol];
            b.y = proj1[(size_t)(krow + 1) * EMBED_DIM + ncol];
            c = __builtin_amdgcn_wmma_f32_16x16x4_f32(
                    false, a1[k], false, b, (short)0, c, false, false);
        }
        // block 2: K = 8 -> 2 chunks of 4
#pragma unroll
        for (int k = 0; k < 2; ++k) {
            const int krow = 4 * k + khalf;
            v2f b;
            b.x = proj2[(size_t)krow * EMBED_DIM + ncol];
            b.y = proj2[(size_t)(krow + 1) * EMBED_DIM + ncol];
            c = __builtin_amdgcn_wmma_f32_16x16x4_f32(
                    false, a2[k], false, b, (short)0, c, false, false);
        }

        // ---- Add block-0 direct gather and store ----
        // C layout: VGPR r, lanes 0-15 -> row M=r; lanes 16-31 -> row M=r+8;
        // column = ncol for all.
#pragma unroll
        for (int r = 0; r < 8; ++r) {
            const int M = r + ((lane >> 4) << 3);
            // lanes 0-15 and 16-31 both loaded rows 0-15's indices, so lane M
            // of the wave holds row M's raw index.
            const int rawM  = __shfl(raw, M);
            const bool in0M = rawM < B0_HI;
            const float m0  = in0M ? 1.0f : 0.0f;
            const int  r0   = in0M ? rawM : 0;
            const float f0  = firstblock_w[(size_t)r0 * EMBED_DIM + ncol] * m0;
            out[(size_t)(tokenBase + M) * EMBED_DIM + ncol] = c[r] + f0;
        }
    }
}

extern "C" void kernel_launch(void* const* d_in, const int* in_sizes, int n_in,
                              void* d_out, int out_size, void* d_ws, size_t ws_size,
                              hipStream_t stream) {
    const int*   inputs       = (const int*)  d_in[0];
    const float* firstblock_w = (const float*)d_in[1];
    const float* emb1         = (const float*)d_in[2];
    const float* proj1        = (const float*)d_in[3];
    const float* emb2         = (const float*)d_in[4];
    const float* proj2        = (const float*)d_in[5];
    float*       out          = (float*)d_out;

    const int tiles  = TOKENS / 16;      // 51200 waves
    const int blocks = tiles / 8;        // 8 waves (256 threads) per block
    blockwise_embed_wmma<<<blocks, 256, 0, stream>>>(
        inputs, firstblock_w, emb1, proj1, emb2, proj2, out);
}